// TransitionDown_85177791414435
// MI455X (gfx1250) — compile-verified
//
#include <hip/hip_runtime.h>
#include <math.h>

// Problem constants from the reference setup
#define B_    4
#define N_    8192
#define M_    2048      // N/STRIDE
#define K_    16        // K_NEAREST
#define INCH  64
#define CCH   68        // 4 + INCH
#define OUTCH 128
#define NTOT  (B_*M_*K_)   // 131072 rows for BN stats

typedef float v2f __attribute__((ext_vector_type(2)));
typedef float v8f __attribute__((ext_vector_type(8)));

// ---------------------------------------------------------------------------
// 1) Farthest point sampling: one block per batch, serial over m iterations.
//    Points + running min-dist live in registers (8 per thread).
// ---------------------------------------------------------------------------
__global__ __launch_bounds__(1024) void fps_kernel(const float* __restrict__ events,
                                                   int* __restrict__ down_idx)
{
    const int b   = blockIdx.x;
    const int tid = threadIdx.x;
    const float* evb = events + (size_t)b * N_ * 4;

    float px[8], py[8], pz[8], dist[8];
#pragma unroll
    for (int t = 0; t < 8; ++t) {
        int j = tid + t * 1024;
        float4 e = ((const float4*)evb)[j];
        px[t] = e.x; py[t] = e.y; pz[t] = e.z;
        dist[t] = 1.0e10f;
    }

    __shared__ float swv[32];
    __shared__ int   swi[32];
    __shared__ int   s_last;

    int last = 0;
    if (tid == 0) down_idx[b * M_ + 0] = 0;   // reference seeds index 0

    for (int i = 1; i < M_; ++i) {
        float lx = evb[last * 4 + 0];
        float ly = evb[last * 4 + 1];
        float lz = evb[last * 4 + 2];

        float bestv = -1.0f; int besti = 0;
#pragma unroll
        for (int t = 0; t < 8; ++t) {
            float dx = px[t] - lx, dy = py[t] - ly, dz = pz[t] - lz;
            float d  = dx * dx + dy * dy + dz * dz;
            dist[t]  = fminf(dist[t], d);
            int j = tid + t * 1024;
            if (dist[t] > bestv || (dist[t] == bestv && j < besti)) { bestv = dist[t]; besti = j; }
        }
        // wave32 argmax butterfly (lowest index wins ties, like jnp.argmax)
#pragma unroll
        for (int off = 16; off >= 1; off >>= 1) {
            float ov = __shfl_xor(bestv, off, 32);
            int   oi = __shfl_xor(besti, off, 32);
            if (ov > bestv || (ov == bestv && oi < besti)) { bestv = ov; besti = oi; }
        }
        int wv = tid >> 5, ln = tid & 31;
        if (ln == 0) { swv[wv] = bestv; swi[wv] = besti; }
        __syncthreads();
        if (wv == 0) {
            float v2 = swv[ln]; int i2 = swi[ln];
#pragma unroll
            for (int off = 16; off >= 1; off >>= 1) {
                float ov = __shfl_xor(v2, off, 32);
                int   oi = __shfl_xor(i2, off, 32);
                if (ov > v2 || (ov == v2 && oi < i2)) { v2 = ov; i2 = oi; }
            }
            if (ln == 0) { s_last = i2; down_idx[b * M_ + i] = i2; }
        }
        __syncthreads();
        last = s_last;
    }
}

// ---------------------------------------------------------------------------
// 2) KNN: one thread per query, LDS tiling of the point cloud (1024 pts/tile).
//    Top-16 kept as a sorted register array (strict < matches top_k ties).
// ---------------------------------------------------------------------------
__global__ __launch_bounds__(256) void knn_kernel(const float* __restrict__ events,
                                                  const int* __restrict__ down_idx,
                                                  int* __restrict__ knn_idx)
{
    const int q  = blockIdx.x * 256 + threadIdx.x;  // 0 .. B_*M_-1 (blocks never straddle batches)
    const int b  = q >> 11;
    const float* evb = events + (size_t)b * N_ * 4;

    const int p = down_idx[q];
    float4 qp = ((const float4*)evb)[p];

    float bd[16]; int bi[16];
#pragma unroll
    for (int t = 0; t < 16; ++t) { bd[t] = 3.402823466e+38f; bi[t] = 0; }

    __shared__ float4 tile[1024];
    for (int base = 0; base < N_; base += 1024) {
        __syncthreads();
        for (int t = threadIdx.x; t < 1024; t += 256)
            tile[t] = ((const float4*)evb)[base + t];
        __syncthreads();
        for (int jj = 0; jj < 1024; ++jj) {
            float4 e = tile[jj];
            float dx = e.x - qp.x, dy = e.y - qp.y, dz = e.z - qp.z;
            float d = dx * dx + dy * dy + dz * dz;
            if (d < bd[15]) {
                bd[15] = d; bi[15] = base + jj;
#pragma unroll
                for (int t = 15; t > 0; --t) {
                    if (bd[t] < bd[t - 1]) {
                        float td = bd[t]; bd[t] = bd[t - 1]; bd[t - 1] = td;
                        int   ti = bi[t]; bi[t] = bi[t - 1]; bi[t - 1] = ti;
                    }
                }
            }
        }
    }
#pragma unroll
    for (int t = 0; t < 16; ++t) knn_idx[q * K_ + t] = bi[t];
}

// ---------------------------------------------------------------------------
// 3) down_events gather: out[0 .. B*m*4)
// ---------------------------------------------------------------------------
__global__ __launch_bounds__(256) void gather_down_events_kernel(const float* __restrict__ events,
                                                                 const int* __restrict__ down_idx,
                                                                 float* __restrict__ out)
{
    int i = blockIdx.x * 256 + threadIdx.x;     // (b, mi)
    if (i >= B_ * M_) return;
    int b = i >> 11;
    int p = down_idx[i];
    ((float4*)out)[i] = ((const float4*)(events + (size_t)b * N_ * 4))[p];
}

// ---------------------------------------------------------------------------
// WMMA tile: one wave computes one (b,mi) group's 16 rows x 16 cols of
// h = g @ W using V_WMMA_F32_16X16X4_F32, K = 68 = 17 steps of 4.
// A layout (ISA 7.12.2, 32-bit A 16x4): lane l -> row M=l&15, K=(l>>4)*2+{0,1}.
// B layout (transpose-symmetric):       lane l -> col N=l&15, K=(l>>4)*2+{0,1}.
// C/D: lane l, vgpr v -> row v+(l>>4)*8, col l&15.
// ---------------------------------------------------------------------------
__device__ inline v8f tile_gemm(const float* __restrict__ events,
                                const float* __restrict__ features,
                                const float* __restrict__ W,
                                const int* __restrict__ down_idx,
                                const int* __restrict__ knn_idx,
                                int group, int wave, int lane)
{
    const int lo = lane & 15;
    const int hi = lane >> 4;
    const int b  = group >> 11;
    const int p  = down_idx[group];
    const int j  = knn_idx[group * K_ + lo];     // this lane's A-row = neighbor lo
    const float* evb = events + (size_t)b * N_ * 4;
    const float* fj  = features + (size_t)b * N_ * INCH + (size_t)j * INCH;
    float4 ej = ((const float4*)evb)[j];
    float4 ep = ((const float4*)evb)[p];
    const int ncol = wave * 16 + lo;             // this lane's B/C column

    v8f acc = {};
    {   // k-slice 0: channels 0..3 = grouped_events - down_events
        v2f a;
        a.x = hi ? (ej.z - ep.z) : (ej.x - ep.x);
        a.y = hi ? (ej.w - ep.w) : (ej.y - ep.y);
        int c0 = hi * 2;
        v2f bb;
        bb.x = W[(c0 + 0) * OUTCH + ncol];
        bb.y = W[(c0 + 1) * OUTCH + ncol];
        acc = __builtin_amdgcn_wmma_f32_16x16x4_f32(false, a, false, bb, (short)0, acc, false, false);
    }
#pragma unroll
    for (int s = 1; s < 17; ++s) {               // k-slices 1..16: gathered features
        int c0 = s * 4 + hi * 2;
        v2f a = *(const v2f*)(fj + (c0 - 4));
        v2f bb;
        bb.x = W[(c0 + 0) * OUTCH + ncol];
        bb.y = W[(c0 + 1) * OUTCH + ncol];
        acc = __builtin_amdgcn_wmma_f32_16x16x4_f32(false, a, false, bb, (short)0, acc, false, false);
    }
    return acc;
}

// ---------------------------------------------------------------------------
// 4) Pass A: GEMM + per-channel sum / sumsq (wave-reduced, then atomics)
// ---------------------------------------------------------------------------
__global__ __launch_bounds__(256) void gemm_stats_kernel(const float* __restrict__ events,
                                                         const float* __restrict__ features,
                                                         const float* __restrict__ W,
                                                         const float* __restrict__ bias,
                                                         const int* __restrict__ down_idx,
                                                         const int* __restrict__ knn_idx,
                                                         float* __restrict__ sum,
                                                         float* __restrict__ sumsq)
{
    const int group = blockIdx.x;
    const int wave  = threadIdx.x >> 5;
    const int lane  = threadIdx.x & 31;
    v8f acc = tile_gemm(events, features, W, down_idx, knn_idx, group, wave, lane);

    const int lo = lane & 15, hi = lane >> 4;
    const int ncol = wave * 16 + lo;
    const float bv = bias[ncol];
    float ls = 0.0f, lq = 0.0f;
#pragma unroll
    for (int v = 0; v < 8; ++v) {
        float h = acc[v] + bv;
        ls += h; lq += h * h;
    }
    ls += __shfl_xor(ls, 16, 32);   // fold the two row-halves of the column
    lq += __shfl_xor(lq, 16, 32);
    if (hi == 0) {
        atomicAdd(sum + ncol, ls);
        atomicAdd(sumsq + ncol, lq);
    }
}

__global__ void zero_stats_kernel(float* __restrict__ s)
{
    s[threadIdx.x] = 0.0f;          // 256 threads: sum[128] + sumsq[128]
}

__global__ void stats_finalize_kernel(const float* __restrict__ sum,
                                      const float* __restrict__ sumsq,
                                      float* __restrict__ mu,
                                      float* __restrict__ istd)
{
    int i = threadIdx.x;
    if (i < OUTCH) {
        float m = sum[i] * (1.0f / (float)NTOT);
        float v = sumsq[i] * (1.0f / (float)NTOT) - m * m;   // biased var, like jnp.var
        mu[i]   = m;
        istd[i] = rsqrtf(v + 1e-5f);
    }
}

// ---------------------------------------------------------------------------
// 5) Pass B: recompute GEMM, BN-normalize, ReLU, max over K (= the 16 tile
//    rows, folded via lane^16 shuffle), write down_features.
// ---------------------------------------------------------------------------
__global__ __launch_bounds__(256) void gemm_out_kernel(const float* __restrict__ events,
                                                       const float* __restrict__ features,
                                                       const float* __restrict__ W,
                                                       const float* __restrict__ bias,
                                                       const float* __restrict__ gamma,
                                                       const float* __restrict__ beta,
                                                       const int* __restrict__ down_idx,
                                                       const int* __restrict__ knn_idx,
                                                       const float* __restrict__ mu,
                                                       const float* __restrict__ istd,
                                                       float* __restrict__ out_feat)
{
    const int group = blockIdx.x;
    const int wave  = threadIdx.x >> 5;
    const int lane  = threadIdx.x & 31;
    v8f acc = tile_gemm(events, features, W, down_idx, knn_idx, group, wave, lane);

    const int lo = lane & 15, hi = lane >> 4;
    const int ncol = wave * 16 + lo;
    const float bv = bias[ncol];
    const float m_ = mu[ncol];
    const float is = istd[ncol];
    const float ga = gamma[ncol];
    const float be = beta[ncol];

    float mx = 0.0f;                 // relu output is >= 0, so 0 is a safe identity
#pragma unroll
    for (int v = 0; v < 8; ++v) {
        float h = (acc[v] + bv - m_) * is * ga + be;
        h = fmaxf(h, 0.0f);
        mx = fmaxf(mx, h);
    }
    mx = fmaxf(mx, __shfl_xor(mx, 16, 32));
    if (hi == 0) out_feat[(size_t)group * OUTCH + ncol] = mx;
}

// ---------------------------------------------------------------------------
// Launch
// ---------------------------------------------------------------------------
extern "C" void kernel_launch(void* const* d_in, const int* in_sizes, int n_in,
                              void* d_out, int out_size, void* d_ws, size_t ws_size,
                              hipStream_t stream)
{
    const float* events   = (const float*)d_in[0];  // [B,N,4]
    const float* features = (const float*)d_in[1];  // [B,N,64]
    const float* W        = (const float*)d_in[2];  // [68,128]
    const float* bias     = (const float*)d_in[3];  // [128]
    const float* gamma    = (const float*)d_in[4];  // [128]
    const float* beta     = (const float*)d_in[5];  // [128]
    float* out = (float*)d_out;                     // [B*m*4] ++ [B*m*128]

    char* ws = (char*)d_ws;
    int*   down_idx = (int*)(ws);                   //  32 KB
    int*   knn_idx  = (int*)(ws + 32768);           // 512 KB
    float* sum      = (float*)(ws + 557056);        // 128
    float* sumsq    = sum + 128;
    float* mu       = sum + 256;
    float* istd     = sum + 384;

    fps_kernel<<<B_, 1024, 0, stream>>>(events, down_idx);
    knn_kernel<<<(B_ * M_) / 256, 256, 0, stream>>>(events, down_idx, knn_idx);
    gather_down_events_kernel<<<(B_ * M_) / 256, 256, 0, stream>>>(events, down_idx, out);
    zero_stats_kernel<<<1, 256, 0, stream>>>(sum);
    gemm_stats_kernel<<<B_ * M_, 256, 0, stream>>>(events, features, W, bias,
                                                   down_idx, knn_idx, sum, sumsq);
    stats_finalize_kernel<<<1, 128, 0, stream>>>(sum, sumsq, mu, istd);
    gemm_out_kernel<<<B_ * M_, 256, 0, stream>>>(events, features, W, bias, gamma, beta,
                                                 down_idx, knn_idx, mu, istd,
                                                 out + (size_t)B_ * M_ * 4);
}